// BlockSampleFixed_47090021434001
// MI455X (gfx1250) — compile-verified
//
#include <hip/hip_runtime.h>
#include <stdint.h>

// ---------------------------------------------------------------------------
// Reference is a one-hot-filter conv == pure patch extraction (0 FLOPs):
//   out[(b,y,x), k, i, j] = in[b, k, y+i-3, x+j-2]   (zero pad; i==3,j>=2 -> 0)
// Shapes: in (16,64,64,64) f32, out (16*64*64, 64, 4, 4) f32 = 256 MiB.
// Bandwidth-bound: ~272 MiB traffic -> ~12 us floor at 23.3 TB/s.
// CDNA5 path: async global->LDS staging (ASYNCcnt) + LDS gather + b128 stores.
// ---------------------------------------------------------------------------

#define C 64
#define H 64
#define W 64
#define BATCH 16

#define ROW_PITCH 68                 // floats per (k,i) LDS row: 64 data + 4 pad (16B-aligned rows)
#define K_PITCH   (4 * ROW_PITCH)    // 272 floats per channel (4 rows)
#define LDS_WORDS (16 * K_PITCH)     // 4352 floats = 17408 bytes

// GLOBAL_LOAD_ASYNC_TO_LDS_B128 via inline asm (builtin uses CUDA-LangAS
// pointer params that have no HIP source spelling). VDST = 32-bit LDS byte
// offset (low 32 bits of flat shared address), VADDR = 64-bit global address.
__device__ __forceinline__ void async_ld_b128(const float* g, float* l) {
  unsigned int       loff = (unsigned int)(uintptr_t)l;       // LDS byte offset
  unsigned long long ga   = (unsigned long long)(uintptr_t)g; // global address
  asm volatile("global_load_async_to_lds_b128 %0, %1, off"
               :: "v"(loff), "v"(ga) : "memory");
}

__device__ __forceinline__ void wait_async0() {
#if __has_builtin(__builtin_amdgcn_s_wait_asynccnt)
  __builtin_amdgcn_s_wait_asynccnt(0);
#else
  asm volatile("s_wait_asynccnt 0" ::: "memory");
#endif
}

__global__ __launch_bounds__(256)
void patch_extract_kernel(const float* __restrict__ in, float* __restrict__ out) {
  __shared__ float lds[LDS_WORDS];

  const int tid = threadIdx.x;
  const int kc  = blockIdx.x;  // channel chunk 0..3 (16 channels each)
  const int y   = blockIdx.y;  // 0..63
  const int b   = blockIdx.z;  // 0..15

  // ---- Phase 1: stage rows y-3..y of 16 channels into LDS -----------------
  // 64 (channel,row) lines x 16 segments of 16B = 1024 b128 transfers.
#pragma unroll
  for (int it = 0; it < 4; ++it) {
    int t    = it * 256 + tid;   // 0..1023
    int seg  = t & 15;           // 16B segment within row
    int line = t >> 4;           // 0..63
    int kl   = line >> 2;        // 0..15 channel-local
    int i    = line & 3;         // filter row 0..3
    float* l = &lds[kl * K_PITCH + i * ROW_PITCH + seg * 4];
    int ysrc = y + i - 3;        // upper bound always <= 63
    if (ysrc >= 0) {
      const float* g = in + ((((size_t)b * C + (kc * 16 + kl)) * H + ysrc) * W + seg * 4);
      async_ld_b128(g, l);
    } else {
      l[0] = 0.0f; l[1] = 0.0f; l[2] = 0.0f; l[3] = 0.0f;  // ds_store zero fill
    }
  }
  wait_async0();       // my wave's async LDS writes landed
  __syncthreads();     // everyone's landed (+ ds zero-fills)

  // ---- Phase 2: gather + coalesced b128 stores ----------------------------
  // 4 lanes per output cell: lane owns filter row i, emits one float4 (j=0..3).
  // Consecutive lanes in a wave write a contiguous 512B region.
#pragma unroll 4
  for (int it = 0; it < 16; ++it) {
    int t    = it * 256 + tid;   // 0..4095
    int i    = t & 3;            // filter row
    int cell = t >> 2;           // 0..1023
    int kl   = cell & 15;        // channel-local (fast dim -> contiguous stores)
    int x    = cell >> 4;        // 0..63

    const float* r = &lds[kl * K_PITCH + i * ROW_PITCH];
    int xm2 = (x >= 2)  ? x - 2 : 0;
    int xm1 = (x >= 1)  ? x - 1 : 0;
    int xp1 = (x <= 62) ? x + 1 : 63;

    float4 v;
    v.x = (x >= 2)  ? r[xm2] : 0.0f;   // j=0 -> column x-2
    v.y = (x >= 1)  ? r[xm1] : 0.0f;   // j=1 -> column x-1
    v.z = r[x];                        // j=2 -> column x   (always in range)
    v.w = (x <= 62) ? r[xp1] : 0.0f;   // j=3 -> column x+1
    if (i == 3) { v.z = 0.0f; v.w = 0.0f; }  // masked taps (i==3, j>=2)

    size_t obase = ((((size_t)b * H + y) * W + x) * C + (kc * 16 + kl)) * 16 + i * 4;
    *reinterpret_cast<float4*>(out + obase) = v;  // global_store_b128
  }
}

extern "C" void kernel_launch(void* const* d_in, const int* in_sizes, int n_in,
                              void* d_out, int out_size, void* d_ws, size_t ws_size,
                              hipStream_t stream) {
  const float* in  = (const float*)d_in[0];
  float*       out = (float*)d_out;
  (void)in_sizes; (void)n_in; (void)out_size; (void)d_ws; (void)ws_size;

  dim3 grid(C / 16, H, BATCH);  // (4, 64, 16)
  patch_extract_kernel<<<grid, 256, 0, stream>>>(in, out);
}